// GINReg_add_70592082477428
// MI455X (gfx1250) — compile-verified
//
#include <hip/hip_runtime.h>
#include <cstddef>

// ---------------------------------------------------------------------------
// GIN forward for MI455X (gfx1250, wave32, WMMA).
// Layer kernel: one 256-thread block (8 waves) owns 32 node rows.
// Each wave owns a 16-wide column slab and two 16x16 accumulators, so every
// B (weight) fragment fetched from L2 is reused across two row tiles --
// halves the dominant weight-refetch traffic vs a 16-row block, and the two
// independent WMMA chains overlap the v_wmma RAW latency.
// ---------------------------------------------------------------------------

typedef __attribute__((ext_vector_type(2))) float v2f;
typedef __attribute__((ext_vector_type(8))) float v8f;

#define DDIM 128
#define MB   32    // rows per block
#define XS   132   // padded LDS row stride (132 mod 64 = 4 -> conflict-free A reads)

static __device__ __forceinline__ v8f wmma4(v2f a, v2f b, v8f c) {
  // D = A(16x4 f32) * B(4x16 f32) + C(16x16 f32)
  return __builtin_amdgcn_wmma_f32_16x16x4_f32(false, a, false, b, (short)0, c,
                                               false, false);
}

static __device__ __forceinline__ void atomAddF(float* p, float v) {
  __hip_atomic_fetch_add(p, v, __ATOMIC_RELAXED, __HIP_MEMORY_SCOPE_AGENT);
}

// ---------------------------------------------------------------------------

__global__ __launch_bounds__(256) void zero_f32(float* __restrict__ p, size_t n) {
  size_t i = (size_t)blockIdx.x * blockDim.x + threadIdx.x;
  size_t stride = (size_t)gridDim.x * blockDim.x;
  for (; i < n; i += stride) p[i] = 0.0f;
}

// Edge scatter-add: 2 edges per block, 128 lanes per edge (one 512B row each).
// h rows live in L2 (51MB total), so this is an L2 gather + f32 atomic stream.
__global__ __launch_bounds__(256) void scatter_add(
    const float* __restrict__ h, const int* __restrict__ src,
    const int* __restrict__ dst, float* __restrict__ agg, int nedges) {
  int e = blockIdx.x * 2 + (threadIdx.x >> 7);
  int c = threadIdx.x & 127;
  if (e < nedges) {
    int s = src[e], d = dst[e];
    atomAddF(&agg[(size_t)d * DDIM + c], h[(size_t)s * DDIM + c]);
  }
}

// 32x16(out) = 32x128(A in LDS) @ 128x16(column slab of W from global/L2).
// One B fragment feeds two independent WMMA chains (row tiles 0 and 1).
static __device__ __forceinline__ void gemm_tile2(
    const float* __restrict__ sX, const float* __restrict__ Wg,
    float* __restrict__ sY, const float* __restrict__ bias,
    int n0, int lrow, int lhi) {
  v8f acc0 = {};
  v8f acc1 = {};
#pragma unroll
  for (int k0 = 0; k0 < DDIM; k0 += 4) {
    int ka = k0 + lhi * 2;
    v2f b, a0, a1;
    // B frag: B[k][n0+lrow]; lanes<16 hold K={0,1}, lanes>=16 K={2,3}
    b[0] = Wg[(size_t)(ka) * DDIM + n0 + lrow];
    b[1] = Wg[(size_t)(ka + 1) * DDIM + n0 + lrow];
    // A frags for the two row tiles (ISA 32-bit 16x4 layout)
    a0[0] = sX[lrow * XS + ka];
    a0[1] = sX[lrow * XS + ka + 1];
    a1[0] = sX[(16 + lrow) * XS + ka];
    a1[1] = sX[(16 + lrow) * XS + ka + 1];
    acc0 = wmma4(a0, b, acc0);
    acc1 = wmma4(a1, b, acc1);
  }
  float bv = bias[n0 + lrow];
#pragma unroll
  for (int v = 0; v < 8; v++) {  // C/D layout: VGPR v -> row v + 8*lhi, col = lane%16
    int m = v + lhi * 8;
    sY[m * XS + n0 + lrow]        = acc0[v] + bv;
    sY[(16 + m) * XS + n0 + lrow] = acc1[v] + bv;
  }
}

// Row LayerNorm over 128 cols + ReLU for 32 rows. 8 threads/row, 16 cols each.
// Contains one internal __syncthreads -> must be called by all 256 threads.
static __device__ __forceinline__ void ln_relu32(
    const float* __restrict__ sIn, float* __restrict__ sOut,
    float* __restrict__ sS, float* __restrict__ sQ,
    const float* __restrict__ gamma, const float* __restrict__ beta, int tid) {
  int r = tid >> 3, p = tid & 7;
  float s = 0.0f, q = 0.0f;
#pragma unroll
  for (int j = 0; j < 16; j++) {
    float v = sIn[r * XS + p * 16 + j];
    s += v; q += v * v;
  }
  sS[r * 8 + p] = s;
  sQ[r * 8 + p] = q;
  __syncthreads();
  float S = 0.0f, Q = 0.0f;
#pragma unroll
  for (int j = 0; j < 8; j++) { S += sS[r * 8 + j]; Q += sQ[r * 8 + j]; }
  float mean = S * (1.0f / DDIM);
  float var  = Q * (1.0f / DDIM) - mean * mean;
  float inv  = rsqrtf(var + 1e-5f);
#pragma unroll
  for (int j = 0; j < 16; j++) {
    int c = p * 16 + j;
    float v = (sIn[r * XS + c] - mean) * inv * gamma[c] + beta[c];
    sOut[r * XS + c] = fmaxf(v, 0.0f);
  }
}

__global__ __launch_bounds__(256) void gin_layer(
    const float* __restrict__ h, const float* __restrict__ agg,
    const float* __restrict__ epsArr, int l,
    const float* __restrict__ W1, const float* __restrict__ b1,
    const float* __restrict__ g1, const float* __restrict__ be1,
    const float* __restrict__ W2, const float* __restrict__ b2,
    const float* __restrict__ ng, const float* __restrict__ nb,
    float* __restrict__ hout, int nrows) {
  __shared__ float sX[MB * XS];
  __shared__ float sY[MB * XS];
  __shared__ float sS[MB * 8];
  __shared__ float sQ[MB * 8];

  int tid  = threadIdx.x;
  int lane = tid & 31;
  int wave = tid >> 5;
  int lrow = lane & 15;
  int lhi  = lane >> 4;
  int n0   = wave * 16;
  int row0 = blockIdx.x * MB;

  const size_t po = (size_t)l * DDIM * DDIM;
  const size_t vo = (size_t)l * DDIM;
  const float* W1l = W1 + po;  const float* b1l = b1 + vo;
  const float* g1l = g1 + vo;  const float* be1l = be1 + vo;
  const float* W2l = W2 + po;  const float* b2l = b2 + vo;
  const float* ngl = ng + vo;  const float* nbl = nb + vo;
  float epsv = 1.0f + epsArr[l];

  // x = (1+eps)*h + agg  (per-row, guarded)
  for (int i = tid; i < MB * DDIM; i += 256) {
    int r = i >> 7, c = i & 127;
    int gr = row0 + r;
    float v = 0.0f;
    if (gr < nrows) {
      size_t o = (size_t)gr * DDIM + c;
      v = epsv * h[o] + agg[o];
    }
    sX[r * XS + c] = v;
  }
  __syncthreads();

  gemm_tile2(sX, W1l, sY, b1l, n0, lrow, lhi);   // y1 = x@W1 + b1
  __syncthreads();
  ln_relu32(sY, sX, sS, sQ, g1l, be1l, tid);     // inner LN + relu -> sX
  __syncthreads();
  gemm_tile2(sX, W2l, sY, b2l, n0, lrow, lhi);   // y2 = .@W2 + b2
  __syncthreads();
  ln_relu32(sY, sY, sS, sQ, ngl, nbl, tid);      // outer LN + relu (in place)
  __syncthreads();

  int r = tid >> 3, p = tid & 7;
  int gr = row0 + r;
  if (gr < nrows) {
#pragma unroll
    for (int j = 0; j < 16; j++) {
      int c = p * 16 + j;
      hout[(size_t)gr * DDIM + c] = sY[r * XS + c];
    }
  }
}

// Graph sum-pool. gid is sorted -> run-length accumulate, flush atomics only on
// graph transitions (~1-2 flushes per 64-node chunk instead of 64).
__global__ __launch_bounds__(128) void pool_graph(
    const float* __restrict__ h, const int* __restrict__ gid,
    float* __restrict__ hg, int nrows) {
  int c = threadIdx.x;
  int base = blockIdx.x * 64;
  float acc = 0.0f;
  int cur = -1;
  for (int i = 0; i < 64; i++) {
    int node = base + i;
    if (node >= nrows) break;
    int g = gid[node];
    if (g != cur) {
      if (cur >= 0) atomAddF(&hg[(size_t)cur * DDIM + c], acc);
      cur = g;
      acc = 0.0f;
    }
    acc += h[(size_t)node * DDIM + c];
  }
  if (cur >= 0) atomAddF(&hg[(size_t)cur * DDIM + c], acc);
}

// Readout: [G,144] @ [144,128] (+LN+relu), then dot with fc2W[128] -> [G,1]
__global__ __launch_bounds__(256) void fc_head(
    const float* __restrict__ hg, const float* __restrict__ desc,
    const float* __restrict__ fc1W, const float* __restrict__ fc1b,
    const float* __restrict__ n1g, const float* __restrict__ n1b,
    const float* __restrict__ fc2W, const float* __restrict__ fc2b,
    float* __restrict__ out, int ngraphs) {
  const int K = 144, AS = 148;  // 148 mod 64 = 20 -> conflict-free strided reads
  __shared__ float sA[16 * AS];
  __shared__ float sY[16 * XS];
  __shared__ float sS[16 * 16];
  __shared__ float sQ[16 * 16];

  int tid  = threadIdx.x;
  int lane = tid & 31;
  int wave = tid >> 5;
  int lrow = lane & 15;
  int lhi  = lane >> 4;
  int n0   = wave * 16;
  int row0 = blockIdx.x * 16;

  // A = concat(hg, desc)
  for (int i = tid; i < 16 * K; i += 256) {
    int r = i / K, c = i % K;
    int g = row0 + r;
    float v = 0.0f;
    if (g < ngraphs)
      v = (c < DDIM) ? hg[(size_t)g * DDIM + c] : desc[(size_t)g * 16 + (c - DDIM)];
    sA[r * AS + c] = v;
  }
  __syncthreads();

  v8f accA = {};
  v8f accB = {};
#pragma unroll
  for (int k0 = 0; k0 < K; k0 += 8) {
    int ka = k0 + lhi * 2;
    int kb = k0 + 4 + lhi * 2;
    v2f a0, b0, a1, b1;
    a0[0] = sA[lrow * AS + ka];     a0[1] = sA[lrow * AS + ka + 1];
    a1[0] = sA[lrow * AS + kb];     a1[1] = sA[lrow * AS + kb + 1];
    b0[0] = fc1W[(size_t)(ka) * DDIM + n0 + lrow];
    b0[1] = fc1W[(size_t)(ka + 1) * DDIM + n0 + lrow];
    b1[0] = fc1W[(size_t)(kb) * DDIM + n0 + lrow];
    b1[1] = fc1W[(size_t)(kb + 1) * DDIM + n0 + lrow];
    accA = wmma4(a0, b0, accA);
    accB = wmma4(a1, b1, accB);
  }
#pragma unroll
  for (int v = 0; v < 8; v++) {
    int m = v + lhi * 8;
    sY[m * XS + n0 + lrow] = accA[v] + accB[v] + fc1b[n0 + lrow];
  }
  __syncthreads();

  // LN over 128 cols (16 rows): 16 threads/row, 8 cols each
  {
    int r = tid >> 4, p = tid & 15;
    float s = 0.0f, q = 0.0f;
#pragma unroll
    for (int j = 0; j < 8; j++) {
      float v = sY[r * XS + p * 8 + j];
      s += v; q += v * v;
    }
    sS[r * 16 + p] = s;
    sQ[r * 16 + p] = q;
    __syncthreads();
    float S = 0.0f, Q = 0.0f;
#pragma unroll
    for (int j = 0; j < 16; j++) { S += sS[r * 16 + j]; Q += sQ[r * 16 + j]; }
    float mean = S * (1.0f / DDIM);
    float var  = Q * (1.0f / DDIM) - mean * mean;
    float inv  = rsqrtf(var + 1e-5f);
#pragma unroll
    for (int j = 0; j < 8; j++) {
      int c = p * 8 + j;
      float v = (sY[r * XS + c] - mean) * inv * n1g[c] + n1b[c];
      sY[r * XS + c] = fmaxf(v, 0.0f);
    }
  }
  __syncthreads();

  // final 128 -> 1 projection (C == 1)
  int r = tid >> 4, p = tid & 15;
  float part = 0.0f;
#pragma unroll
  for (int j = 0; j < 8; j++) {
    int c = p * 8 + j;
    part += sY[r * XS + c] * fc2W[c];
  }
  sS[r * 16 + p] = part;
  __syncthreads();
  if (p == 0) {
    float t = 0.0f;
#pragma unroll
    for (int j = 0; j < 16; j++) t += sS[r * 16 + j];
    int g = row0 + r;
    if (g < ngraphs) out[g] = t + fc2b[0];
  }
}

// ---------------------------------------------------------------------------

extern "C" void kernel_launch(void* const* d_in, const int* in_sizes, int n_in,
                              void* d_out, int out_size, void* d_ws, size_t ws_size,
                              hipStream_t stream) {
  const float* h0    = (const float*)d_in[0];
  const float* desc  = (const float*)d_in[1];
  const int*   src   = (const int*)d_in[2];
  const int*   dst   = (const int*)d_in[3];
  const int*   gid   = (const int*)d_in[4];
  const float* W1    = (const float*)d_in[5];
  const float* b1    = (const float*)d_in[6];
  const float* g1    = (const float*)d_in[7];
  const float* be1   = (const float*)d_in[8];
  const float* W2    = (const float*)d_in[9];
  const float* b2    = (const float*)d_in[10];
  const float* eps   = (const float*)d_in[11];
  const float* ng    = (const float*)d_in[12];
  const float* nb    = (const float*)d_in[13];
  const float* fc1W  = (const float*)d_in[14];
  const float* fc1b  = (const float*)d_in[15];
  const float* n1g   = (const float*)d_in[16];
  const float* n1b   = (const float*)d_in[17];
  const float* fc2W  = (const float*)d_in[18];
  const float* fc2b  = (const float*)d_in[19];

  const int N = in_sizes[4];        // gid: [N]
  const int E = in_sizes[2];        // src: [E]
  const int G = in_sizes[1] / 16;   // desc: [G, 16]
  const int L = in_sizes[11];       // eps: [L]

  const size_t ND = (size_t)N * DDIM;
  float* bufA = (float*)d_ws;            // agg / h ping-pong
  float* bufB = bufA + ND;
  float* hg   = bufB + ND;               // [G, 128]

  const int zgrid = 2048;
  const int ginBlocks = (N + MB - 1) / MB;
  const int scBlocks  = (E + 1) / 2;

  const float* hin = h0;
  float* bufs[2] = {bufA, bufB};
  for (int l = 0; l < L; l++) {
    float* aggb = bufs[l & 1];
    zero_f32<<<zgrid, 256, 0, stream>>>(aggb, ND);
    scatter_add<<<scBlocks, 256, 0, stream>>>(hin, src, dst, aggb, E);
    // agg buffer doubles as h_out: each block only reads/writes its own rows.
    gin_layer<<<ginBlocks, 256, 0, stream>>>(hin, aggb, eps, l,
                                             W1, b1, g1, be1, W2, b2, ng, nb,
                                             aggb, N);
    hin = aggb;
  }

  zero_f32<<<64, 256, 0, stream>>>(hg, (size_t)G * DDIM);
  pool_graph<<<(N + 63) / 64, 128, 0, stream>>>(hin, gid, hg, N);
  fc_head<<<(G + 15) / 16, 256, 0, stream>>>(hg, desc, fc1W, fc1b, n1g, n1b,
                                             fc2W, fc2b, (float*)d_out, G);
}